// NetVLADAggregator_5394478923994
// MI455X (gfx1250) — compile-verified
//
#include <hip/hip_runtime.h>

typedef __attribute__((ext_vector_type(2))) float v2f;
typedef __attribute__((ext_vector_type(8))) float v8f;

#define NK 64      // clusters
#define NC 2048    // feature dim
#define NN 1024    // spatial
#define NB 16      // batch

// ---------------------------------------------------------------------------
// Kernel 1: a[b,k,n] = softmax_k( sum_c w[k,c] * x[b,c,n] )
// One wave computes a 64(k) x 16(n) tile: 4 accumulators of V_WMMA_F32_16X16X4_F32
// M = k (4 tiles of 16), N = n (16 cols), GEMM-K = c.
// Softmax over k is in-lane over 32 regs + one shfl_xor(16) half-combine.
// ---------------------------------------------------------------------------
__global__ __launch_bounds__(128) void netvlad_assign(
    const float* __restrict__ x, const float* __restrict__ w, float* __restrict__ a)
{
    const int lane = threadIdx.x & 31;
    const int wid  = threadIdx.x >> 5;
    const int col  = lane & 15;        // M/N index within fragment
    const int half = lane >> 4;        // 0: lanes 0-15, 1: lanes 16-31
    const int koff = half << 1;        // frag K offset: K = koff + j

    const int tile = blockIdx.x * 4 + wid;      // 0 .. NB*NN/16 - 1 (1024)
    const int b    = tile >> 6;                 // 64 n-tiles per batch
    const int n0   = (tile & 63) << 4;

    const float* xb = x + (size_t)b * NC * NN;

    v8f acc[4] = {v8f{}, v8f{}, v8f{}, v8f{}};

    for (int c = 0; c < NC; c += 4) {
        // A-frags from w (K x C, row-major): element (k = t*16+col, c+koff+j) -> b64 per frag
        v2f af0 = *(const v2f*)(w + (size_t)( 0 + col) * NC + c + koff);
        v2f af1 = *(const v2f*)(w + (size_t)(16 + col) * NC + c + koff);
        v2f af2 = *(const v2f*)(w + (size_t)(32 + col) * NC + c + koff);
        v2f af3 = *(const v2f*)(w + (size_t)(48 + col) * NC + c + koff);
        // B-frag from x (C x N): element (c+koff+j, n0+col) -> 2 coalesced b32 loads
        v2f bf;
        bf.x = xb[(size_t)(c + koff + 0) * NN + n0 + col];
        bf.y = xb[(size_t)(c + koff + 1) * NN + n0 + col];

        acc[0] = __builtin_amdgcn_wmma_f32_16x16x4_f32(false, af0, false, bf, (short)0, acc[0], false, false);
        acc[1] = __builtin_amdgcn_wmma_f32_16x16x4_f32(false, af1, false, bf, (short)0, acc[1], false, false);
        acc[2] = __builtin_amdgcn_wmma_f32_16x16x4_f32(false, af2, false, bf, (short)0, acc[2], false, false);
        acc[3] = __builtin_amdgcn_wmma_f32_16x16x4_f32(false, af3, false, bf, (short)0, acc[3], false, false);
    }

    // Softmax over k (64 values) for the fixed column n = n0+col.
    // Lane l holds k = t*16 + r + 8*half for t=0..3, r=0..7; partner lane l^16 holds the rest.
    float mx = -3.402823466e38f;
#pragma unroll
    for (int t = 0; t < 4; ++t)
#pragma unroll
        for (int r = 0; r < 8; ++r) mx = fmaxf(mx, acc[t][r]);
    mx = fmaxf(mx, __shfl_xor(mx, 16, 32));

    float sum = 0.0f;
#pragma unroll
    for (int t = 0; t < 4; ++t)
#pragma unroll
        for (int r = 0; r < 8; ++r) {
            float e = __expf(acc[t][r] - mx);
            acc[t][r] = e;
            sum += e;
        }
    sum += __shfl_xor(sum, 16, 32);
    const float inv = 1.0f / sum;

    // Store a in (B, K, N) layout: lanes 0-15 and 16-31 each write 16 consecutive floats.
    float* ab = a + (size_t)b * NK * NN;
#pragma unroll
    for (int t = 0; t < 4; ++t)
#pragma unroll
        for (int r = 0; r < 8; ++r) {
            int k = t * 16 + r + 8 * half;
            ab[(size_t)k * NN + n0 + col] = acc[t][r] * inv;
        }
}

// ---------------------------------------------------------------------------
// Kernel 2: a_sum[b,k] = sum_n a[b,k,n]  (deterministic serial row sums)
// ---------------------------------------------------------------------------
__global__ __launch_bounds__(256) void netvlad_asum(
    const float* __restrict__ a, float* __restrict__ asum)
{
    int idx = blockIdx.x * 256 + threadIdx.x;   // 0 .. NB*NK-1 (1024)
    const float4* row = (const float4*)(a + (size_t)idx * NN);
    float s = 0.0f;
    for (int i = 0; i < NN / 4; ++i) {
        float4 v = row[i];
        s += (v.x + v.y) + (v.z + v.w);
    }
    asum[idx] = s;
}

// ---------------------------------------------------------------------------
// Kernel 3: vlad[b,k,c] = sum_n a[b,k,n]*x[b,c,n] - a_sum[b,k]*cent[k,c]
// One wave computes a 64(k) x 16(c) tile; GEMM-K = n. Both A (a) and B (x^T)
// fragments are per-lane contiguous b64 loads. x tile is L2-resident (128MB < 192MB).
// ---------------------------------------------------------------------------
__global__ __launch_bounds__(128) void netvlad_vlad(
    const float* __restrict__ x, const float* __restrict__ a,
    const float* __restrict__ asum, const float* __restrict__ cent,
    float* __restrict__ out)
{
    const int lane = threadIdx.x & 31;
    const int wid  = threadIdx.x >> 5;
    const int col  = lane & 15;
    const int half = lane >> 4;
    const int koff = half << 1;

    const int tile = blockIdx.x * 4 + wid;      // 0 .. NB*NC/16 - 1 (2048)
    const int b    = tile >> 7;                 // 128 c-tiles per batch
    const int c0   = (tile & 127) << 4;

    const float* ab   = a + (size_t)b * NK * NN;
    const float* xrow = x + ((size_t)b * NC + c0 + col) * NN;   // this lane's c-row

    v8f acc[4] = {v8f{}, v8f{}, v8f{}, v8f{}};

    for (int n = 0; n < NN; n += 4) {
        // A-frags from a (K x N): element (k = t*16+col, n+koff+j) -> b64 per frag
        v2f af0 = *(const v2f*)(ab + (size_t)( 0 + col) * NN + n + koff);
        v2f af1 = *(const v2f*)(ab + (size_t)(16 + col) * NN + n + koff);
        v2f af2 = *(const v2f*)(ab + (size_t)(32 + col) * NN + n + koff);
        v2f af3 = *(const v2f*)(ab + (size_t)(48 + col) * NN + n + koff);
        // B-frag = x^T (N x C): element (n+koff+j, c0+col) = x[b, c0+col, n+koff+j] -> b64
        v2f bf = *(const v2f*)(xrow + n + koff);

        acc[0] = __builtin_amdgcn_wmma_f32_16x16x4_f32(false, af0, false, bf, (short)0, acc[0], false, false);
        acc[1] = __builtin_amdgcn_wmma_f32_16x16x4_f32(false, af1, false, bf, (short)0, acc[1], false, false);
        acc[2] = __builtin_amdgcn_wmma_f32_16x16x4_f32(false, af2, false, bf, (short)0, acc[2], false, false);
        acc[3] = __builtin_amdgcn_wmma_f32_16x16x4_f32(false, af3, false, bf, (short)0, acc[3], false, false);
    }

    // D element (t, r): k = t*16 + r + 8*half, c = c0 + col. Coalesced stores per half.
    float* ob = out + (size_t)b * NK * NC;
#pragma unroll
    for (int t = 0; t < 4; ++t)
#pragma unroll
        for (int r = 0; r < 8; ++r) {
            int k = t * 16 + r + 8 * half;
            float s  = asum[b * NK + k];
            float cv = cent[(size_t)k * NC + c0 + col];
            ob[(size_t)k * NC + c0 + col] = acc[t][r] - s * cv;
        }
}

// ---------------------------------------------------------------------------
// Kernel 4: per-batch L2 norm of vlad (fixed-order tree -> deterministic)
// ---------------------------------------------------------------------------
__global__ __launch_bounds__(256) void netvlad_norm(
    const float* __restrict__ out, float* __restrict__ invn)
{
    __shared__ float red[256];
    const float4* vb = (const float4*)(out + (size_t)blockIdx.x * NK * NC);
    float s = 0.0f;
    for (int i = threadIdx.x; i < (NK * NC) / 4; i += 256) {
        float4 v = vb[i];
        s += v.x * v.x + v.y * v.y + v.z * v.z + v.w * v.w;
    }
    red[threadIdx.x] = s;
    __syncthreads();
    for (int off = 128; off > 0; off >>= 1) {
        if (threadIdx.x < off) red[threadIdx.x] += red[threadIdx.x + off];
        __syncthreads();
    }
    if (threadIdx.x == 0)
        invn[blockIdx.x] = 1.0f / fmaxf(sqrtf(red[0]), 1e-12f);
}

// ---------------------------------------------------------------------------
// Kernel 5: scale d_out in place by per-batch inv-norm
// ---------------------------------------------------------------------------
__global__ __launch_bounds__(256) void netvlad_scale(
    float* __restrict__ out, const float* __restrict__ invn)
{
    int gid = blockIdx.x * 256 + threadIdx.x;       // float4 index, 524288 total
    int b = gid / ((NK * NC) / 4);
    float sc = invn[b];
    float4* p = (float4*)out;
    float4 v = p[gid];
    v.x *= sc; v.y *= sc; v.z *= sc; v.w *= sc;
    p[gid] = v;
}

extern "C" void kernel_launch(void* const* d_in, const int* in_sizes, int n_in,
                              void* d_out, int out_size, void* d_ws, size_t ws_size,
                              hipStream_t stream) {
    const float* x    = (const float*)d_in[0];   // (B, C, N)
    const float* w    = (const float*)d_in[1];   // (K, C)
    const float* cent = (const float*)d_in[2];   // (K, C)
    float* out = (float*)d_out;                  // (B, K*C)

    float* a    = (float*)d_ws;                  // (B, K, N)  4 MB
    float* asum = a + (size_t)NB * NK * NN;      // (B, K)     4 KB
    float* invn = asum + NB * NK;                // (B,)       64 B

    netvlad_assign<<<(NB * NN / 16) / 4, 128, 0, stream>>>(x, w, a);
    netvlad_asum  <<<(NB * NK) / 256, 256, 0, stream>>>(a, asum);
    netvlad_vlad  <<<(NB * NC / 16) / 4, 128, 0, stream>>>(x, a, asum, cent, out);
    netvlad_norm  <<<NB, 256, 0, stream>>>(out, invn);
    netvlad_scale <<<(NB * NK * NC / 4) / 256, 256, 0, stream>>>(out, invn);
}